// SetAttention_Linear_fla_18605798326481
// MI455X (gfx1250) — compile-verified
//
#include <hip/hip_runtime.h>

// Sizes from the reference
#define B_   4
#define T_   4096
#define E_   1024
#define H_   16
#define D_   64
#define CH_  64
#define NCH_ (T_ / CH_)
#define LS   68   // LDS row stride (floats): 16B-aligned rows, bank = 4*row+col -> conflict-free frags

typedef __attribute__((ext_vector_type(2))) float v2f;
typedef __attribute__((ext_vector_type(8))) float v8f;

__device__ __forceinline__ v8f wmma_k4(v2f a, v2f b, v8f c) {
  return __builtin_amdgcn_wmma_f32_16x16x4_f32(
      false, a, false, b, (short)0, c, false, false);
}

// ---- CDNA5 async copy: global -> LDS, 16 bytes per lane, tracked by ASYNCcnt ----
__device__ __forceinline__ void async_ld16(float* lds_dst, const float* gsrc) {
  unsigned lo = (unsigned)(unsigned long long)(uintptr_t)lds_dst;  // flat LDS addr[31:0] = LDS offset
  asm volatile("global_load_async_to_lds_b128 %0, %1, off"
               :: "v"(lo), "v"(gsrc) : "memory");
}
__device__ __forceinline__ void wait_async_le2() {
  asm volatile("s_wait_asynccnt 0x2" ::: "memory");
}

// C(i16,j16) += A * B, K=64. A row-major [m][k]; Bt stored TRANSPOSED [n][k]. All b64 frags.
__device__ __forceinline__ v8f mm64_AB(const float* __restrict__ A,
                                       const float* __restrict__ Bt,
                                       int i16, int j16, v8f c) {
  const int lane = threadIdx.x & 31, hf = lane >> 4, lm = lane & 15;
  const float* ap = A  + (i16 + lm) * LS + 2 * hf;
  const float* bp = Bt + (j16 + lm) * LS + 2 * hf;
#pragma unroll
  for (int k0 = 0; k0 < 64; k0 += 4) {
    v2f a = *(const v2f*)(ap + k0);
    v2f b = *(const v2f*)(bp + k0);
    c = wmma_k4(a, b, c);
  }
  return c;
}

// C(i16,j16) += A^T * B: A stored [k][m] (K-matrix [t][d]); Bt transposed [n][k].
__device__ __forceinline__ v8f mm64_ATB(const float* __restrict__ A,
                                        const float* __restrict__ Bt,
                                        int i16, int j16, v8f c) {
  const int lane = threadIdx.x & 31, hf = lane >> 4, lm = lane & 15;
  const float* bp = Bt + (j16 + lm) * LS + 2 * hf;
#pragma unroll
  for (int k0 = 0; k0 < 64; k0 += 4) {
    const int ka = k0 + 2 * hf;
    v2f a, b;
    a[0] = A[ka * LS + (i16 + lm)];
    a[1] = A[(ka + 1) * LS + (i16 + lm)];
    b = *(const v2f*)(bp + k0);
    c = wmma_k4(a, b, c);
  }
  return c;
}

// C/D layout: element r at (M = i16 + r + 8*(lane>>4), N = j16 + lane&15)
__device__ __forceinline__ void st_tile(float* Dst, int i16, int j16, v8f c) {
  const int lane = threadIdx.x & 31, hf = lane >> 4, lm = lane & 15;
#pragma unroll
  for (int r = 0; r < 8; ++r) Dst[(i16 + r + 8 * hf) * LS + j16 + lm] = c[r];
}
__device__ __forceinline__ void st_tile_phi(float* Dst, int i16, int j16, v8f c) {
  const int lane = threadIdx.x & 31, hf = lane >> 4, lm = lane & 15;
#pragma unroll
  for (int r = 0; r < 8; ++r) {
    float xv = c[r];
    Dst[(i16 + r + 8 * hf) * LS + j16 + lm] = xv > 0.0f ? xv + 1.0f : expf(xv);
  }
}
// Transposed tile store/load: storage holds M[n][m]
__device__ __forceinline__ void st_tile_T(float* Dst, int i16, int j16, v8f c) {
  const int lane = threadIdx.x & 31, hf = lane >> 4, lm = lane & 15;
#pragma unroll
  for (int r = 0; r < 8; ++r) Dst[(j16 + lm) * LS + i16 + r + 8 * hf] = c[r];
}
__device__ __forceinline__ v8f ld_tile_T(const float* Src, int i16, int j16) {
  const int lane = threadIdx.x & 31, hf = lane >> 4, lm = lane & 15;
  v8f c;
#pragma unroll
  for (int r = 0; r < 8; ++r) c[r] = Src[(j16 + lm) * LS + i16 + r + 8 * hf];
  return c;
}

__global__ __launch_bounds__(512) void linattn_fused_kernel(
    const float* __restrict__ x,  const float* __restrict__ Wq,
    const float* __restrict__ Wk, const float* __restrict__ Wv,
    const float* __restrict__ Wo, float* __restrict__ out) {
  __shared__ __align__(16) float sWqT[D_ * LS], sWkT[D_ * LS], sWvT[D_ * LS], sWoT[D_ * LS];
  __shared__ __align__(16) float sX[2][CH_ * LS];          // async double-buffered x chunk
  __shared__ __align__(16) float sSc[CH_ * LS];            // masked scores
  __shared__ __align__(16) float sQ[CH_ * LS];             // q (phi), later o
  __shared__ __align__(16) float sK[CH_ * LS];             // k (phi), row-major [t][d]
  __shared__ __align__(16) float sVt[D_ * LS];             // v transposed [e][t]
  __shared__ __align__(16) float sSt[D_ * LS];             // state S transposed [e][d]

  const int bh  = blockIdx.x;
  const int b   = bh / H_;
  const int h   = bh % H_;
  const int tid = threadIdx.x;
  const int wv  = tid >> 5;                 // 16 waves, one 16x16 tile each
  const int i16 = (wv >> 2) * 16, j16 = (wv & 3) * 16;
  const int lane = tid & 31, hf = lane >> 4, lm = lane & 15;

  // Stage weights (transposed) once; zero state.  (First in-loop barrier publishes them.)
  for (int i = tid; i < D_ * D_; i += 512) {
    const int r = i >> 6, cc = i & 63;
    sWqT[cc * LS + r] = Wq[i];
    sWkT[cc * LS + r] = Wk[i];
    sWvT[cc * LS + r] = Wv[i];
    sWoT[cc * LS + r] = Wo[i];
    sSt [r * LS + cc] = 0.0f;
  }

  const float* xg = x   + ((size_t)b * T_) * E_ + (size_t)h * D_;
  float*       og = out + ((size_t)b * T_) * E_ + (size_t)h * D_;

  // Async prefetch of chunk `ch` into sX[buf]: 1024 16B transfers, 2 per thread.
  const int f0r = tid >> 4,          f0c = (tid & 15) * 4;          // rows 0..31
  const int f1r = (tid >> 4) + 32,   f1c = f0c;                     // rows 32..63
  auto issue_chunk = [&](int ch, int buf) {
    const size_t tb = (size_t)(ch * CH_);
    async_ld16(&sX[buf][f0r * LS + f0c], xg + (tb + f0r) * E_ + f0c);
    async_ld16(&sX[buf][f1r * LS + f1c], xg + (tb + f1r) * E_ + f1c);
  };

  issue_chunk(0, 0);  // prologue prefetch

  for (int ch = 0; ch < NCH_; ++ch) {
    const int cur = ch & 1;
    // Prefetch next chunk (wraps harmlessly on last iter -> uniform wait<=2).
    issue_chunk((ch + 1) & (NCH_ - 1), cur ^ 1);
    wait_async_le2();            // oldest 2 asyncs (this chunk's data) complete
    __syncthreads();             // barrier 1: x chunk visible; sQ from prev out-proj now writable

    // Stage A: q = phi(x Wq), k = phi(x Wk), v = x Wv (v stored transposed)   [48 wmma]
    {
      v8f c;
      c = v8f{}; c = mm64_AB(sX[cur], sWqT, i16, j16, c); st_tile_phi(sQ, i16, j16, c);
      c = v8f{}; c = mm64_AB(sX[cur], sWkT, i16, j16, c); st_tile_phi(sK, i16, j16, c);
      c = v8f{}; c = mm64_AB(sX[cur], sWvT, i16, j16, c); st_tile_T(sVt, i16, j16, c);
    }
    __syncthreads();             // barrier 2

    // Stage B: scores = tril(Q K^T) -> sSc ; o_inter = Q @ S_old (regs)       [32 wmma]
    v8f o = v8f{};
    {
      v8f c = v8f{};
      c = mm64_AB(sQ, sK, i16, j16, c);      // K row-major IS the transposed B operand
#pragma unroll
      for (int r = 0; r < 8; ++r) if ((j16 + lm) > (i16 + r + 8 * hf)) c[r] = 0.0f;
      st_tile(sSc, i16, j16, c);
      o = mm64_AB(sQ, sSt, i16, j16, o);     // reads S_old (last write: prev chunk)
    }
    __syncthreads();             // barrier 3

    // Stage C: o += scores @ V ; S += K^T V (own transposed tile) ; stage o into sQ  [32 wmma]
    {
      o = mm64_AB(sSc, sVt, i16, j16, o);
      v8f s = ld_tile_T(sSt, i16, j16);
      s = mm64_ATB(sK, sVt, i16, j16, s);
      st_tile_T(sSt, i16, j16, s);
      st_tile(sQ, i16, j16, o);              // q dead (no sQ readers this stage)
    }
    __syncthreads();             // barrier 4

    // Stage D: out = o @ Wo -> global, non-temporal (out never re-read)       [16 wmma]
    {
      v8f f = v8f{};
      f = mm64_AB(sQ, sWoT, i16, j16, f);
#pragma unroll
      for (int r = 0; r < 8; ++r) {
        const int t = ch * CH_ + i16 + r + 8 * hf;
        __builtin_nontemporal_store(f[r], &og[(size_t)t * E_ + j16 + lm]);
      }
    }
    // No end barrier: program order (StageD -> issue -> async-wait -> barrier 1)
    // already separates StageD's sQ reads from next iteration's sQ writes.
  }
}

extern "C" void kernel_launch(void* const* d_in, const int* in_sizes, int n_in,
                              void* d_out, int out_size, void* d_ws, size_t ws_size,
                              hipStream_t stream) {
  (void)in_sizes; (void)n_in; (void)d_ws; (void)ws_size; (void)out_size;
  const float* x  = (const float*)d_in[0];
  const float* Wq = (const float*)d_in[1];
  const float* Wk = (const float*)d_in[2];
  const float* Wv = (const float*)d_in[3];
  const float* Wo = (const float*)d_in[4];
  float* out = (float*)d_out;
  linattn_fused_kernel<<<dim3(B_ * H_), dim3(512), 0, stream>>>(x, Wq, Wk, Wv, Wo, out);
}